// AtomUpdateBlock_72679436583219
// MI455X (gfx1250) — compile-verified
//
#include <hip/hip_runtime.h>

typedef __attribute__((ext_vector_type(2))) float v2f;
typedef __attribute__((ext_vector_type(8))) float v8f;

#define DE 512
#define DA 256
#define DR 16
#define INV_SQRT2 0.70710678118654752440f

__device__ __forceinline__ float silu_f(float v) {
    return v / (1.0f + __expf(-v));
}

// ---------------------------------------------------------------------------
// Zero the edge-aggregation accumulator (ws is poisoned before timing).
// ---------------------------------------------------------------------------
__global__ void zero_kernel(float4* __restrict__ p, long n4) {
    long i = (long)blockIdx.x * blockDim.x + threadIdx.x;
    long stride = (long)gridDim.x * blockDim.x;
    float4 z = {0.f, 0.f, 0.f, 0.f};
    for (; i < n4; i += stride) p[i] = z;
}

// ---------------------------------------------------------------------------
// Fused: bases_emb = basis_rad @ W_rbf ; x = m * bases_emb ;
//        accum[idx_atom] += x   (global f32 atomics)
// 512 threads/block, thread t owns output column t. W_rbf column in registers.
// ---------------------------------------------------------------------------
#define EG 8  // edges staged per barrier
__global__ __launch_bounds__(DE)
void edge_scatter_kernel(const float* __restrict__ m,
                         const float* __restrict__ basis,
                         const int*   __restrict__ idx,
                         const float* __restrict__ W_rbf,
                         float*       __restrict__ accum,
                         int nEdges) {
    __shared__ float sb[EG][DR];
    const int c = threadIdx.x;  // column 0..511

    float w[DR];
#pragma unroll
    for (int r = 0; r < DR; ++r) w[r] = W_rbf[r * DE + c];

    const int nGroups = (nEdges + EG - 1) / EG;
    for (int g = blockIdx.x; g < nGroups; g += gridDim.x) {
        const int e0 = g * EG;
        if (threadIdx.x < EG * DR) {
            int j = threadIdx.x / DR, r = threadIdx.x % DR;
            int e = e0 + j;
            sb[j][r] = (e < nEdges) ? basis[(long)e * DR + r] : 0.0f;
        }
        __syncthreads();
#pragma unroll 1
        for (int j = 0; j < EG; ++j) {
            int e = e0 + j;
            if (e >= nEdges) break;
            float dot = 0.0f;
#pragma unroll
            for (int r = 0; r < DR; ++r) dot = fmaf(sb[j][r], w[r], dot);
            float x = m[(long)e * DE + c] * dot;
            atomicAdd(&accum[(long)idx[e] * DE + c], x);
        }
        __syncthreads();
    }
}

// ---------------------------------------------------------------------------
// fp32 WMMA GEMM, N fixed = 256.  out = f(A[M,K] @ B[K,256])
//   mode 0: out = silu(acc)
//   mode 1: out = (res + silu(acc)) * INV_SQRT2   (res may alias out)
// Block: 256 threads = 8 waves; wave tile 16x64 (4 accumulators, A reused 4x);
// block tile 32 rows x 256 cols.
// A-frag:  lane holds A[m0 + (lane&15)][k + 2*(lane>>4) + {0,1}]
// B-frag:  lane holds B[k + 2*(lane>>4) + {0,1}][n0 + (lane&15)]
// C/D:     lane&15 = col, VGPR j = row (j + 8*(lane>>4))
// ---------------------------------------------------------------------------
__global__ __launch_bounds__(256)
void gemm_kernel(const float* __restrict__ A,
                 const float* __restrict__ B,
                 const float* __restrict__ res,
                 float*       __restrict__ out,
                 int M, int K, int mode) {
    const int lane = threadIdx.x & 31;
    const int wave = threadIdx.x >> 5;
    const int mw   = wave >> 2;        // 0..1
    const int nw   = wave & 3;         // 0..3
    const int m0   = blockIdx.x * 32 + mw * 16;
    const int n0   = nw * 64;
    const int row  = lane & 15;
    const int hi   = lane >> 4;        // 0 or 1
    const int kb   = hi * 2;
    const int col  = lane & 15;

    v8f acc0 = {}, acc1 = {}, acc2 = {}, acc3 = {};

    const int mrow = min(m0 + row, M - 1);   // clamp: keep EXEC all-1s for WMMA
    const float* Arow = A + (long)mrow * K;

#pragma unroll 4
    for (int k = 0; k < K; k += 4) {
        v2f a;
        a.x = Arow[k + kb];
        a.y = Arow[k + kb + 1];
        const float* B0 = B + (long)(k + kb) * DA + n0 + col;
        const float* B1 = B + (long)(k + kb + 1) * DA + n0 + col;
        v2f b0; b0.x = B0[0];  b0.y = B1[0];
        v2f b1; b1.x = B0[16]; b1.y = B1[16];
        v2f b2; b2.x = B0[32]; b2.y = B1[32];
        v2f b3; b3.x = B0[48]; b3.y = B1[48];
        acc0 = __builtin_amdgcn_wmma_f32_16x16x4_f32(false, a, false, b0, (short)0, acc0, false, false);
        acc1 = __builtin_amdgcn_wmma_f32_16x16x4_f32(false, a, false, b1, (short)0, acc1, false, false);
        acc2 = __builtin_amdgcn_wmma_f32_16x16x4_f32(false, a, false, b2, (short)0, acc2, false, false);
        acc3 = __builtin_amdgcn_wmma_f32_16x16x4_f32(false, a, false, b3, (short)0, acc3, false, false);
    }

    v8f accs[4] = {acc0, acc1, acc2, acc3};
#pragma unroll
    for (int t = 0; t < 4; ++t) {
#pragma unroll
        for (int j = 0; j < 8; ++j) {
            int r = m0 + j + 8 * hi;
            if (r < M) {
                long o = (long)r * DA + n0 + t * 16 + col;
                float v = silu_f(accs[t][j]);
                if (mode) v = (res[o] + v) * INV_SQRT2;
                out[o] = v;
            }
        }
    }
}

// ---------------------------------------------------------------------------
extern "C" void kernel_launch(void* const* d_in, const int* in_sizes, int n_in,
                              void* d_out, int out_size, void* d_ws, size_t ws_size,
                              hipStream_t stream) {
    // inputs: h[A,256](unused), m[E,512], basis_rad[E,16], idx_atom[E],
    //         W_rbf[16,512], W_in[512,256], res_W1[3,256,256], res_W2[3,256,256]
    const float* m     = (const float*)d_in[1];
    const float* basis = (const float*)d_in[2];
    const int*   idx   = (const int*)d_in[3];
    const float* W_rbf = (const float*)d_in[4];
    const float* W_in  = (const float*)d_in[5];
    const float* W1    = (const float*)d_in[6];
    const float* W2    = (const float*)d_in[7];

    const int nAtoms = in_sizes[0] / DA;
    const int nEdges = in_sizes[1] / DE;

    // workspace layout: accum [A,512] | xbuf [A,256] | ybuf [A,256]
    float* accum = (float*)d_ws;
    float* xbuf  = accum + (size_t)nAtoms * DE;
    float* ybuf  = xbuf + (size_t)nAtoms * DA;
    float* outp  = (float*)d_out;

    long n4 = (long)nAtoms * DE / 4;
    zero_kernel<<<1024, 256, 0, stream>>>((float4*)accum, n4);

    edge_scatter_kernel<<<2048, DE, 0, stream>>>(m, basis, idx, W_rbf, accum, nEdges);

    const int gemmBlocks = (nAtoms + 31) / 32;
    // x = silu(accum @ W_in)
    gemm_kernel<<<gemmBlocks, 256, 0, stream>>>(accum, W_in, nullptr, xbuf, nAtoms, DE, 0);

    for (int i = 0; i < 3; ++i) {
        const float* w1 = W1 + (size_t)i * DA * DA;
        const float* w2 = W2 + (size_t)i * DA * DA;
        // y = silu(x @ W1[i])
        gemm_kernel<<<gemmBlocks, 256, 0, stream>>>(xbuf, w1, nullptr, ybuf, nAtoms, DA, 0);
        // x = (x + silu(y @ W2[i])) * 1/sqrt(2)   (last layer -> d_out)
        float* dst = (i == 2) ? outp : xbuf;
        gemm_kernel<<<gemmBlocks, 256, 0, stream>>>(ybuf, w2, xbuf, dst, nAtoms, DA, 1);
    }
}